// ChunkAggregator_80109730005681
// MI455X (gfx1250) — compile-verified
//
#include <hip/hip_runtime.h>
#include <hip/hip_bf16.h>

// Problem constants (fixed by the reference's setup_inputs()).
#define B_    4
#define L_    8192
#define V_    32000
#define D_    1024
#define BLK_  128
#define NB_   (L_ / BLK_)          // 64

// Flat f32 output layout: [new_tokens | cat_emb | hist]
#define NT_SIZE (B_ * (NB_ + L_))          // 33024
#define CE_OFF  NT_SIZE
#define CE_SIZE (B_ * NB_ * D_)            // 262144
#define H_OFF   (CE_OFF + CE_SIZE)         // 295168
#define H_SIZE  (B_ * NB_ * V_)            // 8192000
#define TOT     (H_OFF + H_SIZE)           // 8487168
#define T4      (TOT / 4)                  // 2121792 float4 elements
#define NT4     (NT_SIZE / 4)              // 8256
#define CE4_END ((CE_OFF + CE_SIZE) / 4)   // 73792

typedef int v4i_t __attribute__((ext_vector_type(4)));
typedef __attribute__((address_space(1))) v4i_t* gp128_t;  // global b128 ptr
typedef __attribute__((address_space(3))) v4i_t* lp128_t;  // LDS b128 ptr

// One pass over the whole flat output:
//   region 0: new_tokens (tokens as float; first-of-block prefix then raw)
//   region 1: cat_emb    (row gather from cat_embed_W, 16B vector copies)
//   region 2: hist zeros (bulk of the bytes) -> gfx1250 async store-from-LDS
__global__ __launch_bounds__(256) void ca_fill_kernel(const int* __restrict__ tokens,
                                                      const float* __restrict__ catW,
                                                      float* __restrict__ out) {
    __shared__ float4 zbuf[256];
    zbuf[threadIdx.x] = float4{0.f, 0.f, 0.f, 0.f};
    __syncthreads();

    const long stride = (long)gridDim.x * blockDim.x;
    for (long i = (long)blockIdx.x * blockDim.x + threadIdx.x; i < T4; i += stride) {
        if (i >= CE4_END) {
            // ---- hist zero fill: 16B per lane, streamed from LDS zeros ----
#if __has_builtin(__builtin_amdgcn_global_store_async_from_lds_b128)
            __builtin_amdgcn_global_store_async_from_lds_b128(
                (gp128_t)(v4i_t*)(out + 4 * i),
                (lp128_t)(v4i_t*)&zbuf[threadIdx.x],
                0, 0);
#else
            ((float4*)out)[i] = float4{0.f, 0.f, 0.f, 0.f};
#endif
        } else if (i >= NT4) {
            // ---- cat_emb gather: all 4 lanes of this float4 share one row ----
            long e = 4 * i - CE_OFF;
            int  r = (int)(e / D_);          // row in [0, B*NB)
            int  d = (int)(e % D_);          // 16B-aligned column
            int  b = r / NB_;
            int  n = r - b * NB_;
            int  tok = tokens[b * L_ + n * BLK_];
            float4 v = *(const float4*)(catW + (long)tok * D_ + d);
            ((float4*)out)[i] = v;
        } else {
            // ---- new_tokens: tiny region, scalar per element ----
            long e0 = 4 * i;
            float4 v;
            float* vp = &v.x;
#pragma unroll
            for (int k = 0; k < 4; ++k) {
                long e = e0 + k;
                int  b = (int)(e / (NB_ + L_));
                int  j = (int)(e - (long)b * (NB_ + L_));
                int  ti = (j < NB_) ? (b * L_ + j * BLK_) : (b * L_ + (j - NB_));
                vp[k] = (float)tokens[ti];
            }
            ((float4*)out)[i] = v;
        }
    }

#if __has_builtin(__builtin_amdgcn_s_wait_asynccnt)
    __builtin_amdgcn_s_wait_asynccnt(0);
#endif
}

// Scatter: one thread per token, hardware f32 atomic add into its block's row.
// Rows are disjoint across (b, n) blocks, so contention is only intra-block.
__global__ __launch_bounds__(256) void ca_scatter_kernel(const int* __restrict__ tokens,
                                                         float* __restrict__ hist) {
    int g = blockIdx.x * 256 + threadIdx.x;
    if (g >= B_ * L_) return;
    int b = g / L_;
    int l = g - b * L_;
    int n = l / BLK_;
    int t = tokens[g];
    unsafeAtomicAdd(&hist[(long)(b * NB_ + n) * V_ + t], 1.0f);
}

extern "C" void kernel_launch(void* const* d_in, const int* in_sizes, int n_in,
                              void* d_out, int out_size, void* d_ws, size_t ws_size,
                              hipStream_t stream) {
    (void)in_sizes; (void)n_in; (void)out_size; (void)d_ws; (void)ws_size;

    const int*   tokens = (const int*)d_in[0];
    const float* catW   = (const float*)d_in[1];
    // d_in[2] (num_embed_W) is a dead lookup in the reference; unused.
    float* out = (float*)d_out;

    // Fill everything (new_tokens, cat_emb, hist=0) in one wide-store pass.
    ca_fill_kernel<<<4096, 256, 0, stream>>>(tokens, catW, out);

    // Then scatter-add the per-block histogram counts.
    ca_scatter_kernel<<<(B_ * L_ + 255) / 256, 256, 0, stream>>>(tokens, out + H_OFF);
}